// GaussianModel_36043365548589
// MI455X (gfx1250) — compile-verified
//
#include <hip/hip_runtime.h>
#include <hip/hip_bf16.h>

#define VOL_D 256
#define W 26     // reference fixed window per axis
#define WP 32    // padded to 2 WMMA tiles

typedef __attribute__((ext_vector_type(2))) float v2f;
typedef __attribute__((ext_vector_type(8))) float v8f;

// One block (256 threads = 8 waves, wave32) per Gaussian.
// Phase 1: threads 0..95 compute per-axis weights wx/wy/wz (exp evals) + extents.
// Phase 2: wave 0 computes the 32x32 x-y outer-product slab with 4x
//          v_wmma_f32_16x16x4_f32 (rank-1: only K=0 stripe populated).
// Phase 3: all 8 waves scatter contrib = wxy[x][y] * (I*wz[z]) with native
//          fp32 global atomics (inline asm, SADDR form, device scope) over the
//          EXACT in-bounds box (z fastest for coalescing into L2 lines).
__global__ __launch_bounds__(256) void gaussian_splat_kernel(
    const float* __restrict__ centers,
    const float* __restrict__ sigmas,
    const float* __restrict__ intens,
    float* __restrict__ vol, int n)
{
    int g = blockIdx.x;
    if (g >= n) return;

    __shared__ float s_wx[WP];
    __shared__ float s_wy[WP];
    __shared__ float s_wzI[WP];
    __shared__ float s_wxy[32 * 33];   // padded stride to dodge bank conflicts
    __shared__ int   s_min[3];
    __shared__ int   s_n[3];

    const float scale = (float)(VOL_D - 1);       // 255.0
    float sigma = sigmas[g];
    float inten = intens[g];
    float cx = centers[3 * g + 0];
    float cy = centers[3 * g + 1];
    float cz = centers[3 * g + 2];

    float inv2s2 = -0.5f / (sigma * sigma);
    float cutoff = 3.0f * sigma * scale;

    int t = threadIdx.x;

    // ---- Phase 1: per-axis weights + extents (matches reference masking) ----
    if (t < 96) {
        int axis = t >> 5;        // 0:x 1:y 2:z
        int off  = t & 31;
        float c  = (axis == 0) ? cx : ((axis == 1) ? cy : cz);
        float cv = c * scale;

        float lo = cv - cutoff; if (lo < 0.0f) lo = 0.0f;
        int mini = (int)lo;                                   // trunc == floor (nonneg)
        float hi = cv + cutoff; if (hi > scale) hi = scale;
        int maxi = (int)(hi + 1.0f); if (maxi > VOL_D) maxi = VOL_D;
        int next = maxi - mini;
        if (next > W) next = W;
        if (next < 0) next = 0;
        if (off == 0) { s_min[axis] = mini; s_n[axis] = next; }

        float w = 0.0f;
        if (off < next) {                                     // == reference inb mask
            float coord = (float)(mini + off) / scale;
            float d = coord - c;
            w = expf(inv2s2 * d * d);
        }
        if (axis == 0)      s_wx[off]  = w;
        else if (axis == 1) s_wy[off]  = w;
        else                s_wzI[off] = w * inten;           // fold intensity into z
    }
    __syncthreads();

    // ---- Phase 2: wxy = wx (outer) wy via 4x WMMA f32 16x16x4 (wave 0) ----
    if (t < 32) {
        int lane = t;
        // A-matrix 16x4 layout: VGPR0 lanes 0-15 = (M=lane, K=0); K=1..3 zeroed.
        float a0 = (lane < 16) ? s_wx[lane]      : 0.0f;
        float a1 = (lane < 16) ? s_wx[lane + 16] : 0.0f;
        // B-matrix 4x16: K=0 row in lanes 0-15 of VGPR0; other K stripes zeroed.
        float b0 = (lane < 16) ? s_wy[lane]      : 0.0f;
        float b1 = (lane < 16) ? s_wy[lane + 16] : 0.0f;

        int rbase = (lane >> 4) * 8;   // C/D: lanes 16-31 hold rows M+8
        int col   = lane & 15;

        #pragma unroll
        for (int ti = 0; ti < 2; ++ti) {
            v2f A; A.x = ti ? a1 : a0; A.y = 0.0f;
            #pragma unroll
            for (int tj = 0; tj < 2; ++tj) {
                v2f B; B.x = tj ? b1 : b0; B.y = 0.0f;
                v8f C = {};
                v8f D = __builtin_amdgcn_wmma_f32_16x16x4_f32(
                    /*neg_a=*/false, A, /*neg_b=*/false, B,
                    /*c_mod=*/(short)0, C, /*reuse_a=*/false, /*reuse_b=*/false);
                #pragma unroll
                for (int r = 0; r < 8; ++r)
                    s_wxy[(ti * 16 + rbase + r) * 33 + (tj * 16 + col)] = D[r];
            }
        }
    }
    __syncthreads();

    // ---- Phase 3: scatter-add over the exact box, z fastest ----
    int mx = s_min[0], my = s_min[1], mz = s_min[2];
    int nx = s_n[0],   ny = s_n[1],   nz = s_n[2];
    int nyz   = ny * nz;
    int total = nx * nyz;

    for (int l = t; l < total; l += 256) {
        int xi  = l / nyz;
        int rem = l - xi * nyz;
        int yi  = rem / nz;
        int zi  = rem - yi * nz;

        float c = s_wxy[xi * 33 + yi] * s_wzI[zi];
        unsigned int voff =
            ((unsigned int)((mx + xi) * (VOL_D * VOL_D) + (my + yi) * VOL_D + (mz + zi))) << 2;

        // Native CDNA5 fp32 atomic, no-return (STOREcnt), SADDR form:
        //   global_atomic_add_f32 v_offset, v_data, s[base:base+1]
        // Device scope so adds from all WGPs combine coherently in L2.
        asm volatile("global_atomic_add_f32 %0, %1, %2 scope:SCOPE_DEV"
                     :
                     : "v"(voff), "v"(c), "s"(vol)
                     : "memory");
    }
}

extern "C" void kernel_launch(void* const* d_in, const int* in_sizes, int n_in,
                              void* d_out, int out_size, void* d_ws, size_t ws_size,
                              hipStream_t stream) {
    const float* centers = (const float*)d_in[0];  // [N,3]
    const float* sigmas  = (const float*)d_in[1];  // [N]
    const float* intens  = (const float*)d_in[2];  // [N]
    float* vol = (float*)d_out;                    // 256^3 f32
    int n = in_sizes[1];

    // Output must be zeroed every call (harness poisons once, we accumulate).
    hipMemsetAsync(vol, 0, (size_t)out_size * sizeof(float), stream);

    gaussian_splat_kernel<<<n, 256, 0, stream>>>(centers, sigmas, intens, vol, n);
}